// SentiGAT_42734924595327
// MI455X (gfx1250) — compile-verified
//
#include <hip/hip_runtime.h>
#include <math.h>

// ---------------------------------------------------------------------------
// SentiGAT forward for MI455X (gfx1250, wave32, WMMA).
// All GEMMs use V_WMMA_F32_16X16X4_F32 (fp32-exact matrix core path).
// LDS staging uses GLOBAL_LOAD_ASYNC_TO_LDS_B128 (ASYNCcnt) when available.
// ---------------------------------------------------------------------------

#define Bx 512
#define Dx 512
#define NOBJ 12   // rows taken from word_feat (reference's "obj" arg)
#define NWRD 16   // rows taken from object_feat
#define NOW 28

typedef float v2f __attribute__((ext_vector_type(2)));
typedef float v8f __attribute__((ext_vector_type(8)));
typedef int   avi4 __attribute__((vector_size(16)));   // matches builtin param type

#ifdef __has_builtin
#if __has_builtin(__builtin_amdgcn_global_load_async_to_lds_b128)
#define HAVE_ASYNC_B128 1
#endif
#if __has_builtin(__builtin_amdgcn_s_wait_asynccnt)
#define HAVE_WAIT_ASYNC 1
#endif
#endif

__device__ __forceinline__ void stage_b128(const float* __restrict__ g, float* l) {
#ifdef HAVE_ASYNC_B128
  __builtin_amdgcn_global_load_async_to_lds_b128(
      (__attribute__((address_space(1))) avi4*)g,
      (__attribute__((address_space(3))) avi4*)l, 0, 0);
#else
  l[0] = g[0]; l[1] = g[1]; l[2] = g[2]; l[3] = g[3];
#endif
}

__device__ __forceinline__ void wait_async0() {
#ifdef HAVE_ASYNC_B128
#ifdef HAVE_WAIT_ASYNC
  __builtin_amdgcn_s_wait_asynccnt(0);
#else
  asm volatile("s_wait_asynccnt 0x0" ::: "memory");
#endif
#endif
}

__device__ __forceinline__ float lrelu02(float x) { return x > 0.f ? x : 0.2f * x; }

__device__ __forceinline__ float wave_sum(float x) {
#pragma unroll
  for (int off = 16; off > 0; off >>= 1) x += __shfl_down(x, off, 32);
  return x;
}

// ---------------------------------------------------------------------------
// Kernel 1: per-sample OW GAT (heads=2) + obj/word cross attention -> aligned
// grid = B, block = 256 (8 waves). Dynamic LDS: Xs 32x512, Hs 28x1024.
// ---------------------------------------------------------------------------
__global__ void ow_gat_kernel(const float* __restrict__ word_feat,
                              const float* __restrict__ object_feat,
                              const float* __restrict__ ow_W,
                              const float* __restrict__ ow_asrc,
                              const float* __restrict__ ow_adst,
                              const float* __restrict__ ow_b,
                              float* __restrict__ aligned) {
  extern __shared__ float smem[];
  float* Xs = smem;               // 32*512 (rows 28..31 zero) ; reused as U later
  float* Hs = smem + 32 * 512;    // 28*1024  (node, head*512+d)

  __shared__ float s_src[NOW][2], s_dst[NOW][2];
  __shared__ float alphaw[NWRD][2][13];
  __shared__ float S[NOBJ][NWRD];
  __shared__ float attnw[NWRD];

  const int b    = blockIdx.x;
  const int tid  = threadIdx.x;
  const int wv   = tid >> 5;
  const int lane = tid & 31;
  const int half = lane >> 4;
  const int r    = lane & 15;

  // ---- async-stage X = [word_feat[b] (12); object_feat[b] (16); zeros] ----
  for (int q = tid; q < NOW * 512 / 4; q += 256) {     // 3584 B128 transfers
    int idx4 = q * 4;
    int row = idx4 >> 9, col = idx4 & 511;
    const float* g = (row < NOBJ)
        ? &word_feat[((size_t)b * NOBJ + row) * Dx + col]
        : &object_feat[((size_t)b * NWRD + (row - NOBJ)) * Dx + col];
    stage_b128(g, &Xs[idx4]);
  }
  for (int idx = NOW * 512 + tid; idx < 32 * 512; idx += 256) Xs[idx] = 0.f;
  wait_async0();
  __syncthreads();

  // ---- H(28x1024) = X(28x512) @ ow_W(512x1024) via fp32 WMMA ----
  for (int t = wv; t < 2 * 64; t += 8) {       // 2 M-tiles x 64 N-tiles
    int mt = t & 1, nt = t >> 1;
    v8f c = {};
    for (int k0 = 0; k0 < 512; k0 += 4) {
      int ka = k0 + 2 * half;
      v2f a, bv;
      a.x = Xs[(mt * 16 + r) * 512 + ka];
      a.y = Xs[(mt * 16 + r) * 512 + ka + 1];
      bv.x = ow_W[(size_t)ka * 1024 + nt * 16 + r];
      bv.y = ow_W[(size_t)(ka + 1) * 1024 + nt * 16 + r];
      c = __builtin_amdgcn_wmma_f32_16x16x4_f32(false, a, false, bv, (short)0, c,
                                                false, false);
    }
#pragma unroll
    for (int rr = 0; rr < 8; ++rr) {
      int m = mt * 16 + rr + 8 * half;
      if (m < NOW) Hs[(size_t)m * 1024 + nt * 16 + r] = c[rr];
    }
  }
  __syncthreads();

  // ---- attention logits per node/head ----
  for (int p = wv; p < NOW * 2; p += 8) {
    int n = p >> 1, h = p & 1;
    float ps = 0.f, pd = 0.f;
    for (int d = lane; d < Dx; d += 32) {
      float v = Hs[(size_t)n * 1024 + h * 512 + d];
      ps += v * ow_asrc[h * 512 + d];
      pd += v * ow_adst[h * 512 + d];
    }
    ps = wave_sum(ps); pd = wave_sum(pd);
    if (lane == 0) { s_src[n][h] = ps; s_dst[n][h] = pd; }
  }
  __syncthreads();

  // ---- softmax over incoming edges of each word node (12 obj + self) ----
  if (tid < NWRD * 2) {
    int j = tid >> 1, h = tid & 1;
    float e[13], m = -3.4e38f;
    for (int i = 0; i < NOBJ; ++i) { e[i] = lrelu02(s_src[i][h] + s_dst[NOBJ + j][h]); m = fmaxf(m, e[i]); }
    e[12] = lrelu02(s_src[NOBJ + j][h] + s_dst[NOBJ + j][h]); m = fmaxf(m, e[12]);
    float sum = 0.f;
    for (int i = 0; i < 13; ++i) { e[i] = expf(e[i] - m); sum += e[i]; }
    float inv = 1.f / sum;
    for (int i = 0; i < 13; ++i) alphaw[j][h][i] = e[i] * inv;
  }
  __syncthreads();

  // ---- uo (obj nodes: only self loop) and uw (word nodes), head-mean + bias.
  // Reuse Xs region as U: rows 0..11 = uo, rows 12..27 = uw, stride 512.
  float* U = Xs;
  for (int idx = tid; idx < NOBJ * Dx; idx += 256) {
    int i = idx >> 9, d = idx & 511;
    U[i * 512 + d] = 0.5f * (Hs[(size_t)i * 1024 + d] + Hs[(size_t)i * 1024 + 512 + d]) + ow_b[d];
  }
  __syncthreads();
  for (int idx = tid; idx < NWRD * Dx; idx += 256) {
    int j = idx >> 9, d = idx & 511;
    float acc = 0.f;
#pragma unroll
    for (int h = 0; h < 2; ++h) {
      float s = 0.f;
      for (int i = 0; i < NOBJ; ++i) s += alphaw[j][h][i] * Hs[(size_t)i * 1024 + h * 512 + d];
      s += alphaw[j][h][12] * Hs[(size_t)(NOBJ + j) * 1024 + h * 512 + d];
      acc += s;
    }
    U[(NOBJ + j) * 512 + d] = 0.5f * acc + ow_b[d];
  }
  __syncthreads();

  // ---- S = uo @ uw^T (12x16), softmax rows, column-mean, aligned ----
  for (int p = wv; p < NOBJ * NWRD; p += 8) {
    int i = p >> 4, j = p & 15;
    float acc = 0.f;
    for (int d = lane; d < Dx; d += 32) acc += U[i * 512 + d] * U[(NOBJ + j) * 512 + d];
    acc = wave_sum(acc);
    if (lane == 0) S[i][j] = acc;
  }
  __syncthreads();
  if (tid < NOBJ) {
    float m = -3.4e38f;
    for (int j = 0; j < NWRD; ++j) m = fmaxf(m, S[tid][j]);
    float sum = 0.f;
    for (int j = 0; j < NWRD; ++j) { float p = expf(S[tid][j] - m); S[tid][j] = p; sum += p; }
    float inv = 1.f / sum;
    for (int j = 0; j < NWRD; ++j) S[tid][j] *= inv;
  }
  __syncthreads();
  if (tid < NWRD) {
    float s = 0.f;
    for (int i = 0; i < NOBJ; ++i) s += S[i][tid];
    attnw[tid] = s * (1.f / NOBJ);
  }
  __syncthreads();
  for (int d = tid; d < Dx; d += 256) {
    float acc = 0.f;
    for (int j = 0; j < NWRD; ++j) acc += attnw[j] * U[(NOBJ + j) * 512 + d];
    aligned[(size_t)b * Dx + d] = acc;
  }
}

// ---------------------------------------------------------------------------
// Kernel 2: _ind_norm of the 5 features -> xg (B,5,516) (cols 0..512 valid)
// grid = B*5, block = 256
// ---------------------------------------------------------------------------
__global__ void ind_norm_kernel(const float* __restrict__ text,
                                const float* __restrict__ image,
                                const float* __restrict__ imgtxt,
                                const float* __restrict__ face,
                                const float* __restrict__ aligned,
                                float* __restrict__ xg) {
  __shared__ float rs[256], rq[256];
  __shared__ float sh_inv, sh_v;
  int b = blockIdx.x / 5, f = blockIdx.x % 5;
  const float* src = (f == 0) ? text : (f == 1) ? image : (f == 2) ? imgtxt
                     : (f == 3) ? face : aligned;
  src += (size_t)b * Dx;
  int tid = threadIdx.x;
  float s = 0.f, q = 0.f;
  for (int d = tid; d < Dx; d += 256) { float v = src[d]; s += v; q += v * v; }
  rs[tid] = s; rq[tid] = q;
  __syncthreads();
  for (int st = 128; st > 0; st >>= 1) {
    if (tid < st) { rs[tid] += rs[tid + st]; rq[tid] += rq[tid + st]; }
    __syncthreads();
  }
  if (tid == 0) {
    float v = (rs[0] != 0.f) ? 1.f : 0.f;
    float nrm = sqrtf(rq[0] + v * v);
    nrm = fmaxf(nrm, 1e-12f);
    sh_inv = 1.f / nrm; sh_v = v;
  }
  __syncthreads();
  float inv = sh_inv, vv = sh_v;
  float* dst = xg + ((size_t)b * 5 + f) * 516;
  for (int d = tid; d < 516; d += 256)
    dst[d] = (d < Dx) ? src[d] * inv : (d == Dx ? vv * inv : 0.f);
}

// ---------------------------------------------------------------------------
// Kernel 3: graph5 (two fused GAT layers over 5 nodes) -> fused (B, 2560)
// grid = B, block = 256. Dynamic LDS: As 16x516, Hs 16x512, Ys 16x512.
// ---------------------------------------------------------------------------
__global__ void graph5_kernel(const float* __restrict__ xg,
                              const float* __restrict__ g1_W,
                              const float* __restrict__ g1_asrc,
                              const float* __restrict__ g1_adst,
                              const float* __restrict__ g1_b,
                              const float* __restrict__ g2_W,
                              const float* __restrict__ g2_asrc,
                              const float* __restrict__ g2_adst,
                              const float* __restrict__ g2_b,
                              float* __restrict__ fused) {
  extern __shared__ float smem[];
  float* As = smem;                       // 16*516 (rows 5..15 zero)
  float* Hs = smem + 16 * 516;            // 16*512
  float* Ys = smem + 16 * 516 + 16 * 512; // 16*512 (rows 5..15 zero)

  __shared__ float ss[5], sd[5];
  __shared__ float alpha5[5][5];

  const int b    = blockIdx.x;
  const int tid  = threadIdx.x;
  const int wv   = tid >> 5;
  const int lane = tid & 31;
  const int half = lane >> 4;
  const int r    = lane & 15;

  // async-stage the 5x516 node block; zero-pad rows 5..15 of As and Ys
  for (int q = tid; q < 5 * 516 / 4; q += 256) {       // 645 B128 transfers
    int idx4 = q * 4;
    int row = idx4 / 516, col = idx4 % 516;
    stage_b128(&xg[((size_t)b * 5 + row) * 516 + col], &As[idx4]);
  }
  for (int idx = 5 * 516 + tid; idx < 16 * 516; idx += 256) As[idx] = 0.f;
  for (int idx = tid; idx < 16 * 512; idx += 256) {
    if ((idx >> 9) >= 5) Ys[idx] = 0.f;
  }
  wait_async0();
  __syncthreads();

  for (int st = 0; st < 2; ++st) {
    const float* Wg   = st ? g2_W    : g1_W;
    const float* av   = st ? g2_asrc : g1_asrc;
    const float* ad   = st ? g2_adst : g1_adst;
    const float* bias = st ? g2_b    : g1_b;
    const float* Ap   = st ? Ys : As;
    const int lda  = st ? 512 : 516;
    const int Kpad = st ? 512 : 516;
    const int Kv   = st ? 512 : 513;

    // H(16x512) = Apad @ Wg  (fp32 WMMA; rows 5..15 of A are zero)
    for (int nt = wv; nt < 32; nt += 8) {
      v8f c = {};
      for (int k0 = 0; k0 < Kpad; k0 += 4) {
        int ka = k0 + 2 * half;
        v2f a, bv;
        a.x = Ap[r * lda + ka];
        a.y = Ap[r * lda + ka + 1];
        bv.x = (ka     < Kv) ? Wg[(size_t)ka * 512 + nt * 16 + r] : 0.f;
        bv.y = (ka + 1 < Kv) ? Wg[(size_t)(ka + 1) * 512 + nt * 16 + r] : 0.f;
        c = __builtin_amdgcn_wmma_f32_16x16x4_f32(false, a, false, bv, (short)0, c,
                                                  false, false);
      }
#pragma unroll
      for (int rr = 0; rr < 8; ++rr) {
        int m = rr + 8 * half;
        if (m < 5) Hs[m * 512 + nt * 16 + r] = c[rr];
      }
    }
    __syncthreads();

    if (wv < 5) {
      float ps = 0.f, pd = 0.f;
      for (int d = lane; d < Dx; d += 32) {
        float v = Hs[wv * 512 + d];
        ps += v * av[d];
        pd += v * ad[d];
      }
      ps = wave_sum(ps); pd = wave_sum(pd);
      if (lane == 0) { ss[wv] = ps; sd[wv] = pd; }
    }
    __syncthreads();
    if (tid < 5) {  // dst node tid, softmax over 5 srcs (fully connected + self)
      float e[5], m = -3.4e38f;
      for (int s = 0; s < 5; ++s) { e[s] = lrelu02(ss[s] + sd[tid]); m = fmaxf(m, e[s]); }
      float sum = 0.f;
      for (int s = 0; s < 5; ++s) { e[s] = expf(e[s] - m); sum += e[s]; }
      float inv = 1.f / sum;
      for (int s = 0; s < 5; ++s) alpha5[tid][s] = e[s] * inv;
    }
    __syncthreads();
    for (int idx = tid; idx < 5 * Dx; idx += 256) {
      int n = idx >> 9, d = idx & 511;
      float acc = bias[d];
      for (int s = 0; s < 5; ++s) acc += alpha5[n][s] * Hs[s * 512 + d];
      if (st == 0) Ys[n * 512 + d] = fmaxf(acc, 0.f);
      else         fused[(size_t)b * 2560 + n * 512 + d] = acc;
    }
    __syncthreads();
  }
}

// ---------------------------------------------------------------------------
// Kernel 4: hid = relu(fused(512x2560) @ m1_W(2560x512) + m1_b)  (WMMA)
// grid = 128, block = 256 (8 waves, 1 tile each of 32x32 tiles)
// ---------------------------------------------------------------------------
__global__ void mlp1_kernel(const float* __restrict__ fused,
                            const float* __restrict__ m1_W,
                            const float* __restrict__ m1_b,
                            float* __restrict__ hid) {
  const int lane = threadIdx.x & 31;
  const int wv   = threadIdx.x >> 5;
  const int half = lane >> 4;
  const int r    = lane & 15;
  const int tile = blockIdx.x * 8 + wv;  // 0..1023
  const int mt = tile >> 5, nt = tile & 31;

  v8f c = {};
  for (int k0 = 0; k0 < 2560; k0 += 4) {
    if ((k0 & 63) == 0 && k0 + 256 < 2560)   // keep the weight stream warm in L2/L0
      __builtin_prefetch(&m1_W[(size_t)(k0 + 256) * 512 + nt * 16], 0, 1);
    int ka = k0 + 2 * half;
    v2f a, bv;
    a.x = fused[(size_t)(mt * 16 + r) * 2560 + ka];
    a.y = fused[(size_t)(mt * 16 + r) * 2560 + ka + 1];
    bv.x = m1_W[(size_t)ka * 512 + nt * 16 + r];
    bv.y = m1_W[(size_t)(ka + 1) * 512 + nt * 16 + r];
    c = __builtin_amdgcn_wmma_f32_16x16x4_f32(false, a, false, bv, (short)0, c,
                                              false, false);
  }
  float bias = m1_b[nt * 16 + r];
#pragma unroll
  for (int rr = 0; rr < 8; ++rr) {
    int m = mt * 16 + rr + 8 * half;
    hid[(size_t)m * 512 + nt * 16 + r] = fmaxf(c[rr] + bias, 0.f);
  }
}

// ---------------------------------------------------------------------------
// Kernel 5: out = hid(512x512) @ m2_W(512x3) + m2_b.  grid = B, block = 64
// ---------------------------------------------------------------------------
__global__ void mlp2_kernel(const float* __restrict__ hid,
                            const float* __restrict__ m2_W,
                            const float* __restrict__ m2_b,
                            float* __restrict__ out) {
  __shared__ float red[3][64];
  int b = blockIdx.x, tid = threadIdx.x;
  float p0 = 0.f, p1 = 0.f, p2 = 0.f;
  for (int d = tid; d < Dx; d += 64) {
    float h = hid[(size_t)b * Dx + d];
    p0 += h * m2_W[d * 3 + 0];
    p1 += h * m2_W[d * 3 + 1];
    p2 += h * m2_W[d * 3 + 2];
  }
  red[0][tid] = p0; red[1][tid] = p1; red[2][tid] = p2;
  __syncthreads();
  for (int st = 32; st > 0; st >>= 1) {
    if (tid < st) {
      red[0][tid] += red[0][tid + st];
      red[1][tid] += red[1][tid + st];
      red[2][tid] += red[2][tid + st];
    }
    __syncthreads();
  }
  if (tid < 3) out[(size_t)b * 3 + tid] = red[tid][0] + m2_b[tid];
}

// ---------------------------------------------------------------------------
extern "C" void kernel_launch(void* const* d_in, const int* in_sizes, int n_in,
                              void* d_out, int out_size, void* d_ws, size_t ws_size,
                              hipStream_t stream) {
  (void)in_sizes; (void)n_in; (void)out_size; (void)ws_size;
  const float* text    = (const float*)d_in[0];
  const float* image   = (const float*)d_in[1];
  const float* imgtxt  = (const float*)d_in[2];
  const float* face    = (const float*)d_in[3];
  const float* word    = (const float*)d_in[4];
  const float* object  = (const float*)d_in[5];
  const float* ow_W    = (const float*)d_in[6];
  const float* ow_asrc = (const float*)d_in[7];
  const float* ow_adst = (const float*)d_in[8];
  const float* ow_b    = (const float*)d_in[9];
  const float* g1_W    = (const float*)d_in[10];
  const float* g1_asrc = (const float*)d_in[11];
  const float* g1_adst = (const float*)d_in[12];
  const float* g1_b    = (const float*)d_in[13];
  const float* g2_W    = (const float*)d_in[14];
  const float* g2_asrc = (const float*)d_in[15];
  const float* g2_adst = (const float*)d_in[16];
  const float* g2_b    = (const float*)d_in[17];
  const float* m1_W    = (const float*)d_in[18];
  const float* m1_b    = (const float*)d_in[19];
  const float* m2_W    = (const float*)d_in[20];
  const float* m2_b    = (const float*)d_in[21];
  float* out = (float*)d_out;

  float* ws      = (float*)d_ws;
  float* aligned = ws;                                  // B*512
  float* xg      = aligned + (size_t)Bx * Dx;           // B*5*516
  float* fused   = xg + (size_t)Bx * 5 * 516;           // B*2560
  float* hid     = fused + (size_t)Bx * 2560;           // B*512

  const size_t lds1 = (size_t)(32 * 512 + 28 * 1024) * sizeof(float);    // 180 KB
  const size_t lds3 = (size_t)(16 * 516 + 2 * 16 * 512) * sizeof(float); // ~96 KB

  ow_gat_kernel<<<Bx, 256, lds1, stream>>>(word, object, ow_W, ow_asrc, ow_adst,
                                           ow_b, aligned);
  ind_norm_kernel<<<Bx * 5, 256, 0, stream>>>(text, image, imgtxt, face, aligned, xg);
  graph5_kernel<<<Bx, 256, lds3, stream>>>(xg, g1_W, g1_asrc, g1_adst, g1_b,
                                           g2_W, g2_asrc, g2_adst, g2_b, fused);
  mlp1_kernel<<<128, 256, 0, stream>>>(fused, m1_W, m1_b, hid);
  mlp2_kernel<<<Bx, 64, 0, stream>>>(hid, m2_W, m2_b, out);
}